// PseudoInverse_71433896067375
// MI455X (gfx1250) — compile-verified
//
#include <hip/hip_runtime.h>
#include <stdint.h>

#define KGRID   1024
#define NIMG    512
#define NMEAS   200000
#define NBATCH  8
#define BETA_KB 14.04f

typedef float v2f __attribute__((ext_vector_type(2)));
typedef float v8f __attribute__((ext_vector_type(8)));

// ---------------------------------------------------------------- I0 (A&S 9.8)
__device__ __forceinline__ float i0f_dev(float z) {
    if (z < 3.75f) {
        float t = z * (1.0f / 3.75f);
        float t2 = t * t;
        return 1.0f + t2 * (3.5156229f + t2 * (3.0899424f + t2 * (1.2067492f +
               t2 * (0.2659732f + t2 * (0.0360768f + t2 * 0.0045813f)))));
    } else {
        float it = 3.75f / z;
        float p = 0.39894228f + it * (0.01328592f + it * (0.00225319f +
                  it * (-0.00157565f + it * (0.00916281f + it * (-0.02057706f +
                  it * (0.02635537f + it * (-0.01647633f + it * 0.00392377f)))))));
        return expf(z) * rsqrtf(z) * p;
    }
}

// ------------------------------------------------- DFT matrix generation
// G[r][n] = exp(+2*pi*i * krow(r) * n / 1024), krow(r) = (r+768) mod 1024
__global__ void gen_dft_kernel(float* __restrict__ Gr, float* __restrict__ Gi,
                               float* __restrict__ Grt, float* __restrict__ Git) {
    int tid = blockIdx.x * blockDim.x + threadIdx.x;
    if (tid >= NIMG * KGRID) return;
    int r = tid >> 10;
    int n = tid & (KGRID - 1);
    int krow = (r + 768) & (KGRID - 1);
    int ph = (krow * n) & (KGRID - 1);            // exact integer phase mod K
    float ang = (float)ph * (6.283185307179586f / 1024.0f);
    float s, c;
    sincosf(ang, &s, &c);
    Gr[r * KGRID + n] = c;
    Gi[r * KGRID + n] = s;
    Grt[n * NIMG + r] = c;
    Git[n * NIMG + r] = s;
}

__global__ void gen_apod_kernel(float* __restrict__ invApod) {
    int i = threadIdx.x;                           // blockDim = 512
    float x = ((float)i - 256.0f) * (1.0f / 1024.0f);
    float pj = 3.14159265358979f * 6.0f * x;
    float t = BETA_KB * BETA_KB - pj * pj;
    float st = sqrtf(fabsf(t));
    float num = (t > 0.0f) ? sinhf(st) : sinf(st);
    float ap = num / fmaxf(st, 1e-6f);
    invApod[i] = 1.0f / ap;
}

// ------------------------------------------------- Kaiser-Bessel gridding
__global__ void gridding_kernel(const float* __restrict__ yr,
                                const float* __restrict__ yi,
                                const float* __restrict__ uv,
                                const float* __restrict__ wgt,
                                float* __restrict__ GR, float* __restrict__ GI) {
    int m = blockIdx.x * blockDim.x + threadIdx.x;
    if (m >= NMEAS) return;
    float u = uv[2 * m + 0];
    float v = uv[2 * m + 1];
    float i0b = i0f_dev(BETA_KB);
    float coef = wgt[m] / (i0b * i0b);             // fold weights & 1/I0(b)^2

    int ix[6], iy[6];
    float wx[6], wy[6];
    {
        float k = u * (1024.0f / 6.283185307179586f);
        float km = floorf(k - 3.0f);
        #pragma unroll
        for (int j = 0; j < 6; ++j) {
            float pt = km + (float)(j + 1);
            float d = (k - pt) * (1.0f / 3.0f);
            float arg = fmaxf(1.0f - d * d, 0.0f);
            wx[j] = i0f_dev(BETA_KB * sqrtf(arg)) * coef;
            ix[j] = ((int)pt) & (KGRID - 1);
        }
    }
    {
        float k = v * (1024.0f / 6.283185307179586f);
        float km = floorf(k - 3.0f);
        #pragma unroll
        for (int j = 0; j < 6; ++j) {
            float pt = km + (float)(j + 1);
            float d = (k - pt) * (1.0f / 3.0f);
            float arg = fmaxf(1.0f - d * d, 0.0f);
            wy[j] = i0f_dev(BETA_KB * sqrtf(arg));
            iy[j] = ((int)pt) & (KGRID - 1);
        }
    }
    float vr[NBATCH], vi[NBATCH];
    #pragma unroll
    for (int b = 0; b < NBATCH; ++b) {
        vr[b] = yr[(size_t)b * NMEAS + m];
        vi[b] = yi[(size_t)b * NMEAS + m];
    }
    for (int jx = 0; jx < 6; ++jx) {
        int rowoff = ix[jx] << 10;
        for (int jy = 0; jy < 6; ++jy) {
            float w2 = wx[jx] * wy[jy];
            int base = rowoff + iy[jy];
            #pragma unroll
            for (int b = 0; b < NBATCH; ++b) {
                atomicAdd(GR + (size_t)b * (KGRID * KGRID) + base, vr[b] * w2);
                atomicAdd(GI + (size_t)b * (KGRID * KGRID) + base, vi[b] * w2);
            }
        }
    }
}

// ------------------------------------------------- WMMA complex GEMM
// Computes Cr = Ar*Br - Ai*Bi  (and Ci = Ar*Bi + Ai*Br when WRITE_IM).
#define TM 128
#define TN 64
#define TK 32
#define LDA_S 36
#define LDB_S 80

__device__ __forceinline__ void async_b128(unsigned ldsOff, const void* g) {
    asm volatile("global_load_async_to_lds_b128 %0, %1, off"
                 :: "v"(ldsOff), "v"(g) : "memory");
}
__device__ __forceinline__ void wait_async0() {
    asm volatile("s_wait_asynccnt 0" ::: "memory");
}

template <bool WRITE_IM, bool APOD>
__global__ __launch_bounds__(256)
void gemm_cplx(const float* Ar_, const float* Ai_, int lda, size_t sA,
               const float* Br_, const float* Bi_, int ldb, size_t sB,
               float* Cr_, float* Ci_, int ldc, size_t sC,
               int Kdim, const float* __restrict__ invApod) {
    __shared__ float lA[2][TM * LDA_S];
    __shared__ float lB[2][TK * LDB_S];

    const int bz = blockIdx.z;
    const float* Ar = Ar_ + (size_t)bz * sA;
    const float* Ai = Ai_ + (size_t)bz * sA;
    const float* Br = Br_ + (size_t)bz * sB;
    const float* Bi = Bi_ + (size_t)bz * sB;
    float* Cr = Cr_ + (size_t)bz * sC;
    float* Ci = Ci_ ? (Ci_ + (size_t)bz * sC) : nullptr;

    const int bm = blockIdx.y * TM;
    const int bn = blockIdx.x * TN;
    const int tid = threadIdx.x;
    const int lane = tid & 31;
    const int wid = tid >> 5;           // 8 waves: 4 (M) x 2 (N)
    const int wm = wid >> 1;
    const int wn = wid & 1;
    const int lm = lane & 15;           // fragment row/col within 16
    const int kh = (lane >> 4) << 1;    // K offset: 0 or 2

    const unsigned ldsAr = (unsigned)(size_t)&lA[0][0];
    const unsigned ldsAi = (unsigned)(size_t)&lA[1][0];
    const unsigned ldsBr = (unsigned)(size_t)&lB[0][0];
    const unsigned ldsBi = (unsigned)(size_t)&lB[1][0];

    v8f accR[2][2] = {};
    v8f accI[2][2] = {};

    #pragma unroll 1
    for (int k0 = 0; k0 < Kdim; k0 += TK) {
        // ---- async fill: A tile 128x32 (two planes), B tile 32x64 (two planes)
        #pragma unroll
        for (int i = 0; i < 4; ++i) {
            int idx = tid + i * 256;
            int row = idx >> 3, c4 = idx & 7;
            size_t goff = (size_t)(bm + row) * lda + k0 + c4 * 4;
            unsigned loff = (unsigned)((row * LDA_S + c4 * 4) * 4);
            async_b128(ldsAr + loff, Ar + goff);
            async_b128(ldsAi + loff, Ai + goff);
        }
        #pragma unroll
        for (int i = 0; i < 2; ++i) {
            int idx = tid + i * 256;
            int row = idx >> 4, c4 = idx & 15;
            size_t goff = (size_t)(k0 + row) * ldb + bn + c4 * 4;
            unsigned loff = (unsigned)((row * LDB_S + c4 * 4) * 4);
            async_b128(ldsBr + loff, Br + goff);
            async_b128(ldsBi + loff, Bi + goff);
        }
        if (k0 + TK < Kdim) {   // prefetch next K-tile into L2
            __builtin_prefetch(Ar + (size_t)(bm + (tid >> 3)) * lda + (k0 + TK) + (tid & 7) * 4, 0, 1);
            __builtin_prefetch(Br + (size_t)(k0 + TK + (tid >> 4)) * ldb + bn + (tid & 15) * 4, 0, 1);
        }
        wait_async0();
        __syncthreads();

        // ---- compute: 8 k-steps of 16x16x4 f32 WMMA
        #pragma unroll
        for (int ks = 0; ks < TK; ks += 4) {
            v2f ar[2], ai[2], br[2], bi[2], nbi[2];
            #pragma unroll
            for (int mi = 0; mi < 2; ++mi) {
                int rowl = wm * 32 + mi * 16 + lm;
                ar[mi] = *(const v2f*)&lA[0][rowl * LDA_S + ks + kh];
                ai[mi] = *(const v2f*)&lA[1][rowl * LDA_S + ks + kh];
            }
            #pragma unroll
            for (int ni = 0; ni < 2; ++ni) {
                int coll = wn * 32 + ni * 16 + lm;
                br[ni].x = lB[0][(ks + kh) * LDB_S + coll];
                br[ni].y = lB[0][(ks + kh + 1) * LDB_S + coll];
                bi[ni].x = lB[1][(ks + kh) * LDB_S + coll];
                bi[ni].y = lB[1][(ks + kh + 1) * LDB_S + coll];
                nbi[ni].x = -bi[ni].x;
                nbi[ni].y = -bi[ni].y;
            }
            #pragma unroll
            for (int mi = 0; mi < 2; ++mi)
            #pragma unroll
            for (int ni = 0; ni < 2; ++ni) {
                accR[mi][ni] = __builtin_amdgcn_wmma_f32_16x16x4_f32(
                    false, ar[mi], false, br[ni], (short)0, accR[mi][ni], false, false);
                accR[mi][ni] = __builtin_amdgcn_wmma_f32_16x16x4_f32(
                    false, ai[mi], false, nbi[ni], (short)0, accR[mi][ni], false, false);
                if constexpr (WRITE_IM) {
                    accI[mi][ni] = __builtin_amdgcn_wmma_f32_16x16x4_f32(
                        false, ar[mi], false, bi[ni], (short)0, accI[mi][ni], false, false);
                    accI[mi][ni] = __builtin_amdgcn_wmma_f32_16x16x4_f32(
                        false, ai[mi], false, br[ni], (short)0, accI[mi][ni], false, false);
                }
            }
        }
        __syncthreads();
    }

    // ---- epilogue: C/D layout is M = r + 8*(lane>>4), N = lane&15
    const int khi = lane >> 4;
    #pragma unroll
    for (int mi = 0; mi < 2; ++mi)
    #pragma unroll
    for (int ni = 0; ni < 2; ++ni) {
        #pragma unroll
        for (int r = 0; r < 8; ++r) {
            int gm = bm + wm * 32 + mi * 16 + r + 8 * khi;
            int gn = bn + wn * 32 + ni * 16 + lm;
            float vR = accR[mi][ni][r];
            if constexpr (APOD) vR *= invApod[gm] * invApod[gn];
            Cr[(size_t)gm * ldc + gn] = vR;
            if constexpr (WRITE_IM) Ci[(size_t)gm * ldc + gn] = accI[mi][ni][r];
        }
    }
}

// ----------------------------------------------------------------- launch
extern "C" void kernel_launch(void* const* d_in, const int* in_sizes, int n_in,
                              void* d_out, int out_size, void* d_ws, size_t ws_size,
                              hipStream_t stream) {
    const float* y_real  = (const float*)d_in[0];  // (8, 200000)
    const float* y_imag  = (const float*)d_in[1];  // (8, 200000)
    const float* uv      = (const float*)d_in[2];  // (200000, 2)
    const float* weights = (const float*)d_in[3];  // (200000,)
    float* out = (float*)d_out;                    // (8, 512, 512)

    // workspace layout (floats)
    float* GR  = (float*)d_ws;                                   // 8 x 1024^2
    float* GI  = GR  + (size_t)NBATCH * KGRID * KGRID;           // 8 x 1024^2
    float* TR  = GI  + (size_t)NBATCH * KGRID * KGRID;           // 8 x 1024x512
    float* TI  = TR  + (size_t)NBATCH * KGRID * NIMG;            // 8 x 1024x512
    float* Gr  = TI  + (size_t)NBATCH * KGRID * NIMG;            // 512x1024
    float* Gi  = Gr  + (size_t)NIMG * KGRID;                     // 512x1024
    float* Grt = Gi  + (size_t)NIMG * KGRID;                     // 1024x512
    float* Git = Grt + (size_t)KGRID * NIMG;                     // 1024x512
    float* invApod = Git + (size_t)KGRID * NIMG;                 // 512

    // zero the accumulation grids (must happen every call: atomics accumulate)
    hipMemsetAsync(GR, 0, (size_t)2 * NBATCH * KGRID * KGRID * sizeof(float), stream);

    gen_dft_kernel<<<(NIMG * KGRID) / 256, 256, 0, stream>>>(Gr, Gi, Grt, Git);
    gen_apod_kernel<<<1, NIMG, 0, stream>>>(invApod);

    gridding_kernel<<<(NMEAS + 255) / 256, 256, 0, stream>>>(
        y_real, y_imag, uv, weights, GR, GI);

    // Stage A: T = grid * G^T   (1024x512 per batch, full complex)
    gemm_cplx<true, false><<<dim3(NIMG / TN, KGRID / TM, NBATCH), 256, 0, stream>>>(
        GR, GI, KGRID, (size_t)KGRID * KGRID,
        Grt, Git, NIMG, (size_t)0,
        TR, TI, NIMG, (size_t)KGRID * NIMG,
        KGRID, invApod);

    // Stage B: out = Re(G * T) * deapod  (512x512 per batch)
    gemm_cplx<false, true><<<dim3(NIMG / TN, NIMG / TM, NBATCH), 256, 0, stream>>>(
        Gr, Gi, KGRID, (size_t)0,
        TR, TI, NIMG, (size_t)KGRID * NIMG,
        out, nullptr, NIMG, (size_t)NIMG * NIMG,
        KGRID, invApod);

    (void)in_sizes; (void)n_in; (void)out_size; (void)ws_size;
}